// VoxelResBackBone8x_7773890806393
// MI455X (gfx1250) — compile-verified
//
#include <hip/hip_runtime.h>

// ---------------------------------------------------------------------------
// VoxelResBackBone8x for MI455X (gfx1250), fp32 end-to-end.
// Convs: implicit-GEMM on V_WMMA_F32_16X16X4_F32, NDHWC activations, weights
// repacked to (Cout, taps, Cin). Out-of-bounds taps read a zeroed guard buffer
// so A-loads are unconditional (no EXEC juggling in the hot loop). Each wave
// computes NT cout-tiles per A-fragment (NT=4 for C>=64) to amortize gathers.
// ---------------------------------------------------------------------------

typedef float v2f __attribute__((ext_vector_type(2)));
typedef float v8f __attribute__((ext_vector_type(8)));

#define BN_EPS 1e-3f

// ---------------- conv3d: implicit GEMM on fp32 WMMA (NDHWC) ----------------
// grid.x covers 16-position tiles (4 waves / 128-thread block),
// grid.y = Cout/(16*NT). wr layout: (Cout, KD*KH*KW, Cin) flat.
template <int KD, int KH, int KW, int NT>
__global__ __launch_bounds__(128) void conv3d_wmma_ndhwc(
    const float* __restrict__ x, const float* __restrict__ wr,
    const float* __restrict__ bias, float* __restrict__ y,
    const float* __restrict__ zguard,
    int N, int Cin, int Din, int Hin, int Win,
    int Cout, int Dout, int Hout, int Wout,
    int sd, int sh, int sw, int pd, int ph, int pw)
{
    const int lane = threadIdx.x & 31;
    const int wave = threadIdx.x >> 5;
    const int wavesPerBlock = blockDim.x >> 5;
    const int P = N * Dout * Hout * Wout;
    const int numTiles = (P + 15) >> 4;
    const int tile = blockIdx.x * wavesPerBlock + wave;
    if (tile >= numTiles) return;            // wave-uniform exit, EXEC stays full
    const int coutBase = blockIdx.y * (16 * NT);

    // A-fragment row owned by this lane (ISA 7.12.2: 32-bit A 16x4 layout)
    const int m = lane & 15;
    const int pos = tile * 16 + m;
    const bool pv = pos < P;
    int t = pv ? pos : 0;
    const int xo = t % Wout; t /= Wout;
    const int yo = t % Hout; t /= Hout;
    const int zo = t % Dout; const int n = t / Dout;
    const int zb = zo * sd - pd, yb = yo * sh - ph, xb = xo * sw - pw;

    const int Ktot = KD * KH * KW * Cin;
    const int kOff = (lane >> 4) << 1;       // lanes 0-15 -> K%4 in {0,1}; 16-31 -> {2,3}

    const long long sY = (long long)Win * Cin;        // NDHWC strides (floats)
    const long long sZ = (long long)Hin * sY;
    const float* xn = x + (long long)n * Din * sZ;

    // B-fragments: lane m holds wr[coutBase+q*16+m][k0+kOff .. +1]
    const float* wrow[NT];
    #pragma unroll
    for (int q = 0; q < NT; ++q)
        wrow[q] = wr + (long long)(coutBase + q * 16 + m) * Ktot + kOff;

    __builtin_prefetch(wrow[0], 0, 0);

    v8f zero = {};
    v8f acc[NT];
    #pragma unroll
    for (int q = 0; q < NT; ++q) acc[q] = zero;

    #pragma unroll
    for (int dk = 0; dk < KD; ++dk) {
        const int zi = zb + dk;
        const bool zok = (unsigned)zi < (unsigned)Din;
        #pragma unroll
        for (int hk = 0; hk < KH; ++hk) {
            const int yi = yb + hk;
            const bool yok = zok && ((unsigned)yi < (unsigned)Hin);
            #pragma unroll
            for (int wk = 0; wk < KW; ++wk) {
                const int xi = xb + wk;
                const bool ok = pv && yok && ((unsigned)xi < (unsigned)Win);
                // zero-guard: OOB taps read zeros; load is unconditional.
                const float* ap = ok
                    ? (xn + (long long)zi * sZ + (long long)yi * sY
                          + (long long)xi * Cin + kOff)
                    : (zguard + kOff);
                #pragma unroll 4
                for (int c0 = 0; c0 < Cin; c0 += 4) {
                    v2f a = *(const v2f*)ap;          // two adjacent channels: b64
                    #pragma unroll
                    for (int q = 0; q < NT; ++q) {
                        v2f b = *(const v2f*)wrow[q]; // two adjacent K weights: b64
                        // (neg_a, A, neg_b, B, c_mod, C, reuse_a, reuse_b)
                        acc[q] = __builtin_amdgcn_wmma_f32_16x16x4_f32(
                            false, a, false, b, (short)0, acc[q], false, false);
                        wrow[q] += 4;
                    }
                    ap += 4;
                }
            }
        }
    }

    // D layout: VGPR r -> M = r + (lane>=16 ? 8 : 0), N = lane&15.
    // NDHWC store: y[pos*Cout + co]; lanes 0-15 cover 16 adjacent channels.
    const int col = lane & 15;
    const int rowBase = (lane >> 4) << 3;
    #pragma unroll
    for (int q = 0; q < NT; ++q) {
        union { v8f v; float f[8]; } u; u.v = acc[q];
        const float bv = bias ? bias[coutBase + q * 16 + col] : 0.0f;
        #pragma unroll
        for (int r = 0; r < 8; ++r) {
            int p2 = tile * 16 + rowBase + r;
            if (p2 < P)
                y[(long long)p2 * Cout + coutBase + q * 16 + col] = u.f[r] + bv;
        }
    }
}

// ---------------- weight repack: (Cout,Cin,kdim) -> (Cout,kdim,Cin) ----------------
__global__ __launch_bounds__(256) void repack_w_kernel(
    const float* __restrict__ w, float* __restrict__ wr,
    int Cout, int Cin, int kdim)
{
    int i = blockIdx.x * blockDim.x + threadIdx.x;
    int tot = Cout * Cin * kdim;
    if (i >= tot) return;
    int ci  = i % Cin;
    int t   = i / Cin;
    int tap = t % kdim;
    int co  = t / kdim;
    wr[i] = w[((long long)co * Cin + ci) * kdim + tap];
}

// ---------------- scatter voxels -> dense (NDHWC, C=4) + mask ----------------
__global__ __launch_bounds__(256) void scatter_kernel(
    const float* __restrict__ vf, const int* __restrict__ coords,
    float* __restrict__ dense, float* __restrict__ mask,
    int nvox, int D, int H, int W)
{
    int i = blockIdx.x * blockDim.x + threadIdx.x;
    if (i >= nvox) return;
    int b  = coords[4 * i + 0];
    int z  = coords[4 * i + 1];
    int yy = coords[4 * i + 2];
    int xx = coords[4 * i + 3];
    long long sp = (long long)D * H * W;
    long long p  = (long long)b * sp + ((long long)z * H + yy) * W + xx;
    const float4 v = *(const float4*)(vf + 4 * i);
    *(float4*)(dense + 4 * p) = v;
    mask[p] = 1.0f;
}

// ---------------- mask downsample: any-in-window ----------------
__global__ __launch_bounds__(256) void mask_down_kernel(
    const float* __restrict__ min_, float* __restrict__ mout,
    int N, int Di, int Hi, int Wi, int Do_, int Ho_, int Wo_,
    int kd, int kh, int kw, int sd, int sh, int sw, int pd, int ph, int pw)
{
    int i = blockIdx.x * blockDim.x + threadIdx.x;
    int tot = N * Do_ * Ho_ * Wo_;
    if (i >= tot) return;
    int t = i;
    int xo = t % Wo_; t /= Wo_;
    int yo = t % Ho_; t /= Ho_;
    int zo = t % Do_; int n = t / Do_;
    float v = 0.0f;
    for (int a = 0; a < kd; ++a) {
        int zi = zo * sd - pd + a;
        if ((unsigned)zi >= (unsigned)Di) continue;
        for (int b = 0; b < kh; ++b) {
            int yi = yo * sh - ph + b;
            if ((unsigned)yi >= (unsigned)Hi) continue;
            for (int c = 0; c < kw; ++c) {
                int xi = xo * sw - pw + c;
                if ((unsigned)xi >= (unsigned)Wi) continue;
                if (min_[((long long)n * Di + zi) * Hi * Wi + (long long)yi * Wi + xi] > 0.0f)
                    v = 1.0f;
            }
        }
    }
    mout[i] = v;
}

// ---------------- mask voxel count ----------------
__global__ __launch_bounds__(256) void mask_count_kernel(
    const float* __restrict__ mask, int total, float* __restrict__ out)
{
    float s = 0.0f;
    for (int i = blockIdx.x * blockDim.x + threadIdx.x; i < total; i += gridDim.x * blockDim.x)
        s += mask[i];
    __shared__ float sh[256];
    sh[threadIdx.x] = s; __syncthreads();
    for (int off = 128; off > 0; off >>= 1) {
        if (threadIdx.x < off) sh[threadIdx.x] += sh[threadIdx.x + off];
        __syncthreads();
    }
    if (threadIdx.x == 0) atomicAdd(out, sh[0]);
}

// ---------------- masked BN: coalesced sum over NDHWC ----------------
// blockDim = 256 and C | 256, so (linear index & (C-1)) is constant per thread:
// each thread privately accumulates one channel, cross-reduces in LDS, then
// 2 global atomicAdds per channel per block.
__global__ __launch_bounds__(256) void bn_sum_kernel(
    const float* __restrict__ x, const float* __restrict__ mask,
    float* __restrict__ sums, long long total /*N*sp*/, int Cm1, int lc)
{
    const int tid = threadIdx.x;
    const long long nTot = total << lc;
    float s1 = 0.0f, s2 = 0.0f;
    const long long stride = (long long)gridDim.x * blockDim.x;
    for (long long i = (long long)blockIdx.x * blockDim.x + tid; i < nTot; i += stride) {
        float mv = mask[i >> lc];
        if (mv != 0.0f) { float v = x[i]; s1 += v; s2 += v * v; }
    }
    __shared__ float sh1[256], sh2[256];
    sh1[tid] = s1; sh2[tid] = s2;
    __syncthreads();
    const int C = Cm1 + 1;
    for (int off = 128; off >= C; off >>= 1) {
        if (tid < off) { sh1[tid] += sh1[tid + off]; sh2[tid] += sh2[tid + off]; }
        __syncthreads();
    }
    if (tid < C) {
        atomicAdd(&sums[2 * tid],     sh1[tid]);
        atomicAdd(&sums[2 * tid + 1], sh2[tid]);
    }
}

__global__ void bn_finalize_kernel(const float* __restrict__ sums,
                                   const float* __restrict__ cnt,
                                   float* __restrict__ stats, int C)
{
    int c = threadIdx.x;
    if (c >= C) return;
    float nn = cnt[0]; if (nn < 1.0f) nn = 1.0f;
    float mean = sums[2 * c] / nn;
    float var  = sums[2 * c + 1] / nn - mean * mean;
    if (var < 0.0f) var = 0.0f;
    stats[2 * c]     = mean;
    stats[2 * c + 1] = rsqrtf(var + BN_EPS);
}

// ---------------- masked BN apply (+residual)(+relu)(*mask), NDHWC ----------------
__global__ __launch_bounds__(256) void bn_apply_kernel(
    const float* __restrict__ x, const float* __restrict__ res,
    float* __restrict__ dst, const float* __restrict__ stats,
    const float* __restrict__ g, const float* __restrict__ b,
    const float* __restrict__ mask, long long nTot, int Cm1, int lc, int doRelu)
{
    long long i = (long long)blockIdx.x * blockDim.x + threadIdx.x;
    if (i >= nTot) return;
    int c = (int)(i & Cm1);
    float v = (x[i] - stats[2 * c]) * stats[2 * c + 1] * g[c] + b[c];
    if (res) v += res[i];
    if (doRelu) v = fmaxf(v, 0.0f);
    v *= mask[i >> lc];
    dst[i] = v;
}

// Final head: read NDHWC, write NCDHW d_out.
__global__ __launch_bounds__(256) void bn_apply_out_kernel(
    const float* __restrict__ x, float* __restrict__ dst,
    const float* __restrict__ stats,
    const float* __restrict__ g, const float* __restrict__ b,
    const float* __restrict__ mask, long long nTot, int Cm1, int lc, int spo, int C)
{
    long long i = (long long)blockIdx.x * blockDim.x + threadIdx.x;
    if (i >= nTot) return;
    int c = (int)(i & Cm1);
    long long p = i >> lc;
    float v = (x[i] - stats[2 * c]) * stats[2 * c + 1] * g[c] + b[c];
    v = fmaxf(v, 0.0f);
    v *= mask[p];
    int n = (int)(p / spo), s = (int)(p % spo);
    dst[((long long)n * C + c) * spo + s] = v;
}

// ---------------- host orchestration ----------------

struct CBN { const float *w, *g, *b; };
struct BB  { const float *w1, *b1, *g1, *be1, *w2, *b2, *g2, *be2; };

extern "C" void kernel_launch(void* const* d_in, const int* in_sizes, int n_in,
                              void* d_out, int out_size, void* d_ws, size_t ws_size,
                              hipStream_t stream)
{
    (void)out_size;
    constexpr int NB = 2;
    constexpr int NVOX = 80000;
    constexpr int D1=25,H1=160,W1=176, C1=16;
    constexpr int D2=13,H2=80, W2=88,  C2=32;
    constexpr int D3=7, H3=40, W3=44,  C3=64;
    constexpr int D4=3, H4=20, W4=22,  C4=128;
    constexpr int Do=1, Ho=20, Wo=22,  Co=128;
    constexpr int SP1 = D1*H1*W1, SP2 = D2*H2*W2, SP3 = D3*H3*W3, SP4 = D4*H4*W4, SPo = Do*Ho*Wo;

    constexpr size_t SZ1    = (size_t)NB * C1 * SP1 * sizeof(float);   // 90,112,000 B
    constexpr size_t M1_OFF = 3 * SZ1;
    constexpr size_t M1_SZ  = (size_t)NB * SP1 * sizeof(float);
    constexpr size_t M2_OFF = M1_OFF + M1_SZ;
    constexpr size_t M2_SZ  = ((size_t)NB * SP2 * sizeof(float) + 255) / 256 * 256;
    constexpr size_t M3_OFF = M2_OFF + M2_SZ;
    constexpr size_t M3_SZ  = ((size_t)NB * SP3 * sizeof(float) + 255) / 256 * 256;
    constexpr size_t M4_OFF = M3_OFF + M3_SZ;
    constexpr size_t M4_SZ  = ((size_t)NB * SP4 * sizeof(float) + 255) / 256 * 256;
    constexpr size_t MO_OFF = M4_OFF + M4_SZ;
    constexpr size_t MO_SZ  = ((size_t)NB * SPo * sizeof(float) + 255) / 256 * 256;
    constexpr size_t CNT_OFF  = MO_OFF + MO_SZ;       // 5 floats, padded to 256
    constexpr size_t STAT_OFF = CNT_OFF + 256;        // 128 * {mean, invstd}
    constexpr size_t SUM_OFF  = STAT_OFF + 1024;      // 128 * {s1, s2}
    constexpr size_t WR_OFF   = SUM_OFF + 1024;       // repacked weights (max 442368 f)
    constexpr size_t WR_SZ    = 442368 * sizeof(float);
    constexpr size_t ZG_OFF   = WR_OFF + WR_SZ;       // zero guard for OOB taps
    constexpr size_t ZG_SZ    = 1024;
    constexpr size_t WS_NEED  = ZG_OFF + ZG_SZ;
    if (ws_size < WS_NEED) return;

    char*  ws   = (char*)d_ws;
    float* bufA = (float*)(ws);
    float* bufB = (float*)(ws + SZ1);
    float* bufC = (float*)(ws + 2 * SZ1);
    float* m1   = (float*)(ws + M1_OFF);
    float* m2   = (float*)(ws + M2_OFF);
    float* m3   = (float*)(ws + M3_OFF);
    float* m4   = (float*)(ws + M4_OFF);
    float* mo   = (float*)(ws + MO_OFF);
    float* cnts = (float*)(ws + CNT_OFF);
    float* stats= (float*)(ws + STAT_OFF);
    float* sums = (float*)(ws + SUM_OFF);
    float* wrb  = (float*)(ws + WR_OFF);
    float* zg   = (float*)(ws + ZG_OFF);

    // ---- parameter pointer mapping (auto-detect flatten order) ----
    auto f = [&](int i) { return (const float*)d_in[i]; };
    const float* vf = nullptr; const int* coords = nullptr;
    CBN Pin{}, Pd2{}, Pd3{}, Pd4{}, Pout{};
    BB  R1[2], R2[2], R3[2], R4[2];

    bool insertion = (n_in >= 81) && (in_sizes[0] == NVOX * 4) && (in_sizes[1] == 16 * 4 * 27);
    if (insertion) {
        vf = f(0); coords = (const int*)d_in[80];
        auto bb = [&](int i) { return BB{ f(i), f(i+1), f(i+2), f(i+3), f(i+4), f(i+5), f(i+6), f(i+7) }; };
        Pin  = { f(1),  f(2),  f(3)  };
        R1[0] = bb(4);  R1[1] = bb(12);
        Pd2  = { f(20), f(21), f(22) };
        R2[0] = bb(23); R2[1] = bb(31);
        Pd3  = { f(39), f(40), f(41) };
        R3[0] = bb(42); R3[1] = bb(50);
        Pd4  = { f(58), f(59), f(60) };
        R4[0] = bb(61); R4[1] = bb(69);
        Pout = { f(77), f(78), f(79) };
    } else {
        // jax pytree sorted-key order: batch, params(sorted), coords, vf
        vf = f(81); coords = (const int*)d_in[80];
        Pd2 = { f(3),  f(2),  f(1)  };
        Pd3 = { f(6),  f(5),  f(4)  };
        Pd4 = { f(9),  f(8),  f(7)  };
        Pin = { f(12), f(11), f(10) };
        Pout= { f(15), f(14), f(13) };
        auto bbS = [&](int i) { return BB{ f(i+6), f(i+0), f(i+4), f(i+2), f(i+7), f(i+1), f(i+5), f(i+3) }; };
        R1[0] = bbS(16); R1[1] = bbS(24);
        R2[0] = bbS(32); R2[1] = bbS(40);
        R3[0] = bbS(48); R3[1] = bbS(56);
        R4[0] = bbS(64); R4[1] = bbS(72);
    }

    // ---- launch helpers ----
    auto run_conv = [&](const float* xin, const float* wt, const float* bias, float* yout,
                        int Cin, int Di, int Hi, int Wi, int Cout, int Dd, int Hh, int Ww,
                        int kd, int kh, int kw, int sd, int sh, int sw, int pd, int ph, int pw) {
        int kdim = kd * kh * kw;
        int wtot = Cout * Cin * kdim;
        repack_w_kernel<<<(wtot + 255) / 256, 256, 0, stream>>>(wt, wrb, Cout, Cin, kdim);
        int P = NB * Dd * Hh * Ww;
        int tiles = (P + 15) / 16;
        if (kd == 3 && kh == 3 && kw == 3) {
            if (Cout == 16) {
                dim3 gr((tiles + 3) / 4, Cout / 16);
                conv3d_wmma_ndhwc<3,3,3,1><<<gr, 128, 0, stream>>>(
                    xin, wrb, bias, yout, zg, NB, Cin, Di, Hi, Wi, Cout, Dd, Hh, Ww,
                    sd, sh, sw, pd, ph, pw);
            } else if (Cout == 32) {
                dim3 gr((tiles + 3) / 4, Cout / 32);
                conv3d_wmma_ndhwc<3,3,3,2><<<gr, 128, 0, stream>>>(
                    xin, wrb, bias, yout, zg, NB, Cin, Di, Hi, Wi, Cout, Dd, Hh, Ww,
                    sd, sh, sw, pd, ph, pw);
            } else {
                dim3 gr((tiles + 3) / 4, Cout / 64);
                conv3d_wmma_ndhwc<3,3,3,4><<<gr, 128, 0, stream>>>(
                    xin, wrb, bias, yout, zg, NB, Cin, Di, Hi, Wi, Cout, Dd, Hh, Ww,
                    sd, sh, sw, pd, ph, pw);
            }
        } else {
            dim3 gr((tiles + 3) / 4, Cout / 64);
            conv3d_wmma_ndhwc<3,1,1,4><<<gr, 128, 0, stream>>>(
                xin, wrb, bias, yout, zg, NB, Cin, Di, Hi, Wi, Cout, Dd, Hh, Ww,
                sd, sh, sw, pd, ph, pw);
        }
    };
    auto run_bn = [&](const float* xb, const float* resi, float* dst,
                      const float* g, const float* b, const float* mask, const float* cnt,
                      int C, int spatial, int relu) {
        int lc = __builtin_ctz(C);
        long long total = (long long)NB * spatial;
        long long nTot = total << lc;
        hipMemsetAsync(sums, 0, 2 * C * sizeof(float), stream);
        long long blk = (nTot + 255) / 256; if (blk > 2048) blk = 2048;
        bn_sum_kernel<<<(int)blk, 256, 0, stream>>>(xb, mask, sums, total, C - 1, lc);
        bn_finalize_kernel<<<1, C, 0, stream>>>(sums, cnt, stats, C);
        bn_apply_kernel<<<(int)((nTot + 255) / 256), 256, 0, stream>>>(
            xb, resi, dst, stats, g, b, mask, nTot, C - 1, lc, relu);
    };
    auto run_count = [&](const float* mask, int total, float* cnt) {
        int blocks = (total + 255) / 256; if (blocks > 1024) blocks = 1024;
        mask_count_kernel<<<blocks, 256, 0, stream>>>(mask, total, cnt);
    };
    auto run_mdown = [&](const float* mi, float* mout, int Di, int Hi, int Wi,
                         int Dd, int Hh, int Ww, int kd, int kh, int kw,
                         int sd, int sh, int sw, int pd, int ph, int pw) {
        int tot = NB * Dd * Hh * Ww;
        mask_down_kernel<<<(tot + 255) / 256, 256, 0, stream>>>(
            mi, mout, NB, Di, Hi, Wi, Dd, Hh, Ww, kd, kh, kw, sd, sh, sw, pd, ph, pw);
    };
    auto run_block = [&](float* X, float* T1, float* T2, const BB& P, const float* mask,
                         const float* cnt, int C, int Di, int Hi, int Wi) {
        int sp = Di * Hi * Wi;
        run_conv(X,  P.w1, P.b1, T1, C, Di, Hi, Wi, C, Di, Hi, Wi, 3,3,3, 1,1,1, 1,1,1);
        run_bn(T1, nullptr, T1, P.g1, P.be1, mask, cnt, C, sp, 1);
        run_conv(T1, P.w2, P.b2, T2, C, Di, Hi, Wi, C, Di, Hi, Wi, 3,3,3, 1,1,1, 1,1,1);
        run_bn(T2, X, X, P.g2, P.be2, mask, cnt, C, sp, 1);   // relu(bn+res)*mask -> X
    };

    // ---- forward pass ----
    hipMemsetAsync(bufB, 0, (size_t)NB * 4 * SP1 * sizeof(float), stream);  // dense NDHWC, C=4
    hipMemsetAsync(m1,   0, M1_SZ, stream);
    hipMemsetAsync(cnts, 0, 256, stream);
    hipMemsetAsync(zg,   0, ZG_SZ, stream);

    scatter_kernel<<<(NVOX + 255) / 256, 256, 0, stream>>>(vf, coords, bufB, m1, NVOX, D1, H1, W1);
    run_count(m1, NB * SP1, &cnts[0]);

    // stem: cbr(dense, in, m1)
    run_conv(bufB, Pin.w, nullptr, bufA, 4, D1, H1, W1, C1, D1, H1, W1, 3,3,3, 1,1,1, 1,1,1);
    run_bn(bufA, nullptr, bufA, Pin.g, Pin.b, m1, &cnts[0], C1, SP1, 1);

    float* X = bufA; float* T1 = bufB; float* T2 = bufC;
    run_block(X, T1, T2, R1[0], m1, &cnts[0], C1, D1, H1, W1);
    run_block(X, T1, T2, R1[1], m1, &cnts[0], C1, D1, H1, W1);

    // level 2
    run_mdown(m1, m2, D1, H1, W1, D2, H2, W2, 3,3,3, 2,2,2, 1,1,1);
    run_count(m2, NB * SP2, &cnts[1]);
    run_conv(X, Pd2.w, nullptr, T1, C1, D1, H1, W1, C2, D2, H2, W2, 3,3,3, 2,2,2, 1,1,1);
    run_bn(T1, nullptr, T1, Pd2.g, Pd2.b, m2, &cnts[1], C2, SP2, 1);
    { float* t = X; X = T1; T1 = t; }
    run_block(X, T1, T2, R2[0], m2, &cnts[1], C2, D2, H2, W2);
    run_block(X, T1, T2, R2[1], m2, &cnts[1], C2, D2, H2, W2);

    // level 3
    run_mdown(m2, m3, D2, H2, W2, D3, H3, W3, 3,3,3, 2,2,2, 1,1,1);
    run_count(m3, NB * SP3, &cnts[2]);
    run_conv(X, Pd3.w, nullptr, T1, C2, D2, H2, W2, C3, D3, H3, W3, 3,3,3, 2,2,2, 1,1,1);
    run_bn(T1, nullptr, T1, Pd3.g, Pd3.b, m3, &cnts[2], C3, SP3, 1);
    { float* t = X; X = T1; T1 = t; }
    run_block(X, T1, T2, R3[0], m3, &cnts[2], C3, D3, H3, W3);
    run_block(X, T1, T2, R3[1], m3, &cnts[2], C3, D3, H3, W3);

    // level 4 (pad (0,1,1))
    run_mdown(m3, m4, D3, H3, W3, D4, H4, W4, 3,3,3, 2,2,2, 0,1,1);
    run_count(m4, NB * SP4, &cnts[3]);
    run_conv(X, Pd4.w, nullptr, T1, C3, D3, H3, W3, C4, D4, H4, W4, 3,3,3, 2,2,2, 0,1,1);
    run_bn(T1, nullptr, T1, Pd4.g, Pd4.b, m4, &cnts[3], C4, SP4, 1);
    { float* t = X; X = T1; T1 = t; }
    run_block(X, T1, T2, R4[0], m4, &cnts[3], C4, D4, H4, W4);
    run_block(X, T1, T2, R4[1], m4, &cnts[3], C4, D4, H4, W4);

    // output head: k=(3,1,1), s=(2,1,1), p=0 -> NCDHW d_out
    run_mdown(m4, mo, D4, H4, W4, Do, Ho, Wo, 3,1,1, 2,1,1, 0,0,0);
    run_count(mo, NB * SPo, &cnts[4]);
    run_conv(X, Pout.w, nullptr, T1, C4, D4, H4, W4, Co, Do, Ho, Wo, 3,1,1, 2,1,1, 0,0,0);
    {
        int lc = __builtin_ctz(Co);
        long long total = (long long)NB * SPo;
        long long nTot = total << lc;
        hipMemsetAsync(sums, 0, 2 * Co * sizeof(float), stream);
        long long blk = (nTot + 255) / 256; if (blk > 2048) blk = 2048;
        bn_sum_kernel<<<(int)blk, 256, 0, stream>>>(T1, mo, sums, total, Co - 1, lc);
        bn_finalize_kernel<<<1, Co, 0, stream>>>(sums, &cnts[4], stats, Co);
        bn_apply_out_kernel<<<(int)((nTot + 255) / 256), 256, 0, stream>>>(
            T1, (float*)d_out, stats, Pout.g, Pout.b, mo, nTot, Co - 1, lc, SPo, Co);
    }
}